// GaussianModel_11304353923597
// MI455X (gfx1250) — compile-verified
//
#include <hip/hip_runtime.h>
#include <cstdint>
#include <cstddef>

#define TILE 256

// ---- gfx1250 async global<->LDS path (ASYNCcnt), with safe fallback ----
#if defined(__has_builtin)
#if __has_builtin(__builtin_amdgcn_global_load_async_to_lds_b128) && \
    __has_builtin(__builtin_amdgcn_global_store_async_from_lds_b128) && \
    __has_builtin(__builtin_amdgcn_s_wait_asynccnt)
#define USE_ASYNC_LDS 1
#endif
#endif

// CPol temporal hint: NT (non-temporal) — pure streaming, don't retain in cache.
#define CPOL_NT 1

typedef int v4i __attribute__((ext_vector_type(4)));
typedef __attribute__((address_space(3))) v4i lds_v4i;

#define AS_GLOBAL(p) ((v4i*)(p))
#define AS_LDS(p)    ((lds_v4i*)(p))

// Fast HW math: v_rcp_f32 / v_sqrt_f32 / v_exp_f32 (~1 ulp), fine for this op.
__device__ __forceinline__ float frcp(float x)  { return __builtin_amdgcn_rcpf(x); }
__device__ __forceinline__ float fsqrt(float x) { return __builtin_amdgcn_sqrtf(x); }

// Core per-Gaussian math: cov = R diag(f) R^T, opac = sigmoid(o) * sqrt(det(v)/det(f))
__device__ __forceinline__ void gauss_math(
    float o, float s0, float s1, float s2,
    float qw, float qx, float qy, float qz,
    float w0, float w1, float w2,
    float c[9], float* op_out)
{
    // opacity: sigmoid (finite inputs -> nan_to_num is identity)
    float op = frcp(1.0f + __expf(-o));

    // variances = exp(s)^2 ; filtered = v + w
    float e0 = __expf(s0), e1 = __expf(s1), e2 = __expf(s2);
    float v0 = e0 * e0, v1 = e1 * e1, v2 = e2 * e2;
    float f0 = v0 + w0, f1 = v1 + w1, f2 = v2 + w2;

    // norm_coeff = sqrt(det(v) / det(f))
    float detv = v0 * v1 * v2;
    float detf = f0 * f1 * f2;
    op *= fsqrt(detv * frcp(detf));
    *op_out = op;

    // scales_act^2 == filtered variances (skip sqrt->square round trip)

    // normalize quaternion: q /= max(||q||, 1e-12)
    float nn  = qw * qw + qx * qx + qy * qy + qz * qz;
    float inv = frcp(fmaxf(fsqrt(nn), 1e-12f));
    qw *= inv; qx *= inv; qy *= inv; qz *= inv;

    float xx = qx * qx, yy = qy * qy, zz = qz * qz;
    float xy = qx * qy, xz = qx * qz, yz = qy * qz;
    float wx = qw * qx, wy = qw * qy, wz = qw * qz;

    float r00 = 1.0f - 2.0f * (yy + zz), r01 = 2.0f * (xy - wz), r02 = 2.0f * (xz + wy);
    float r10 = 2.0f * (xy + wz), r11 = 1.0f - 2.0f * (xx + zz), r12 = 2.0f * (yz - wx);
    float r20 = 2.0f * (xz - wy), r21 = 2.0f * (yz + wx), r22 = 1.0f - 2.0f * (xx + yy);

    // cov[i][j] = sum_k R[i][k] * f[k] * R[j][k]  (symmetric)
    float c00 = f0 * r00 * r00 + f1 * r01 * r01 + f2 * r02 * r02;
    float c01 = f0 * r00 * r10 + f1 * r01 * r11 + f2 * r02 * r12;
    float c02 = f0 * r00 * r20 + f1 * r01 * r21 + f2 * r02 * r22;
    float c11 = f0 * r10 * r10 + f1 * r11 * r11 + f2 * r12 * r12;
    float c12 = f0 * r10 * r20 + f1 * r11 * r21 + f2 * r12 * r22;
    float c22 = f0 * r20 * r20 + f1 * r21 * r21 + f2 * r22 * r22;

    c[0] = c00; c[1] = c01; c[2] = c02;
    c[3] = c01; c[4] = c11; c[5] = c12;
    c[6] = c02; c[7] = c12; c[8] = c22;
}

__global__ __launch_bounds__(TILE) void gaussian_cov_kernel(
    const float* __restrict__ opa,   // (N,1)
    const float* __restrict__ scl,   // (N,3)
    const float* __restrict__ rot,   // (N,4)
    const float* __restrict__ flt,   // (N,3)
    float* __restrict__ cov_out,     // (N,3,3)
    float* __restrict__ opa_out,     // (N,1)
    int n)
{
    __shared__ __align__(16) float s_rot[TILE * 4];   // 4 KB
    __shared__ __align__(16) float s_scl[TILE * 3];   // 3 KB
    __shared__ __align__(16) float s_flt[TILE * 3];   // 3 KB
    __shared__ __align__(16) float s_opa[TILE];       // 1 KB
    __shared__ __align__(16) float s_cov[TILE * 9];   // 9 KB
    __shared__ __align__(16) float s_opo[TILE];       // 1 KB

    const int t = threadIdx.x;
    const size_t base = (size_t)blockIdx.x * TILE;

    if (base + TILE <= (size_t)n) {
        // ---------------- load: fully coalesced contiguous b128 ----------------
#ifdef USE_ASYNC_LDS
        __builtin_amdgcn_global_load_async_to_lds_b128(
            AS_GLOBAL(rot + base * 4 + (size_t)t * 4), AS_LDS(s_rot + t * 4), 0, CPOL_NT);
        if (t < 192) {
            __builtin_amdgcn_global_load_async_to_lds_b128(
                AS_GLOBAL(scl + base * 3 + (size_t)t * 4), AS_LDS(s_scl + t * 4), 0, CPOL_NT);
            __builtin_amdgcn_global_load_async_to_lds_b128(
                AS_GLOBAL(flt + base * 3 + (size_t)t * 4), AS_LDS(s_flt + t * 4), 0, CPOL_NT);
        }
        if (t < 64) {
            __builtin_amdgcn_global_load_async_to_lds_b128(
                AS_GLOBAL(opa + base + (size_t)t * 4), AS_LDS(s_opa + t * 4), 0, CPOL_NT);
        }
        __builtin_amdgcn_s_wait_asynccnt(0);
#else
        *reinterpret_cast<float4*>(&s_rot[t * 4]) =
            reinterpret_cast<const float4*>(rot + base * 4)[t];
        if (t < 192) {
            *reinterpret_cast<float4*>(&s_scl[t * 4]) =
                reinterpret_cast<const float4*>(scl + base * 3)[t];
            *reinterpret_cast<float4*>(&s_flt[t * 4]) =
                reinterpret_cast<const float4*>(flt + base * 3)[t];
        }
        if (t < 64) {
            *reinterpret_cast<float4*>(&s_opa[t * 4]) =
                reinterpret_cast<const float4*>(opa + base)[t];
        }
#endif
        __syncthreads();

        // ---------------- compute (odd LDS strides 3/9: bank-conflict free) ----------------
        float4 q = *reinterpret_cast<const float4*>(&s_rot[t * 4]); // ds_load_b128
        float c[9], op;
        gauss_math(s_opa[t],
                   s_scl[t * 3 + 0], s_scl[t * 3 + 1], s_scl[t * 3 + 2],
                   q.x, q.y, q.z, q.w,
                   s_flt[t * 3 + 0], s_flt[t * 3 + 1], s_flt[t * 3 + 2],
                   c, &op);
#pragma unroll
        for (int k = 0; k < 9; ++k) s_cov[t * 9 + k] = c[k];
        s_opo[t] = op;

        __syncthreads();

        // ---------------- store: drain LDS as contiguous b128 streams ----------------
        float* covg = cov_out + base * 9;   // tile is 2304 floats = 576 x 16B chunks
#ifdef USE_ASYNC_LDS
        __builtin_amdgcn_global_store_async_from_lds_b128(
            AS_GLOBAL(covg + (size_t)t * 4), AS_LDS(s_cov + t * 4), 0, CPOL_NT);
        __builtin_amdgcn_global_store_async_from_lds_b128(
            AS_GLOBAL(covg + (size_t)(t + 256) * 4), AS_LDS(s_cov + (t + 256) * 4), 0, CPOL_NT);
        if (t < 64) {
            __builtin_amdgcn_global_store_async_from_lds_b128(
                AS_GLOBAL(covg + (size_t)(t + 512) * 4), AS_LDS(s_cov + (t + 512) * 4), 0, CPOL_NT);
            __builtin_amdgcn_global_store_async_from_lds_b128(
                AS_GLOBAL(opa_out + base + (size_t)t * 4), AS_LDS(s_opo + t * 4), 0, CPOL_NT);
        }
        // S_ENDPGM performs an implicit wait-idle; async stores complete before wave retires.
#else
        reinterpret_cast<float4*>(covg)[t] =
            *reinterpret_cast<const float4*>(&s_cov[t * 4]);
        reinterpret_cast<float4*>(covg)[t + 256] =
            *reinterpret_cast<const float4*>(&s_cov[(t + 256) * 4]);
        if (t < 64) {
            reinterpret_cast<float4*>(covg)[t + 512] =
                *reinterpret_cast<const float4*>(&s_cov[(t + 512) * 4]);
            reinterpret_cast<float4*>(opa_out + base)[t] =
                *reinterpret_cast<const float4*>(&s_opo[t * 4]);
        }
#endif
    } else {
        // -------- tail (never taken for N % 256 == 0): guarded scalar path --------
        size_t idx = base + (size_t)t;
        if (idx < (size_t)n) {
            float c[9], op;
            gauss_math(opa[idx],
                       scl[idx * 3 + 0], scl[idx * 3 + 1], scl[idx * 3 + 2],
                       rot[idx * 4 + 0], rot[idx * 4 + 1], rot[idx * 4 + 2], rot[idx * 4 + 3],
                       flt[idx * 3 + 0], flt[idx * 3 + 1], flt[idx * 3 + 2],
                       c, &op);
#pragma unroll
            for (int k = 0; k < 9; ++k) cov_out[idx * 9 + k] = c[k];
            opa_out[idx] = op;
        }
    }
}

extern "C" void kernel_launch(void* const* d_in, const int* in_sizes, int n_in,
                              void* d_out, int out_size, void* d_ws, size_t ws_size,
                              hipStream_t stream)
{
    (void)n_in; (void)out_size; (void)d_ws; (void)ws_size;
    const float* opa = (const float*)d_in[0];   // (N,1)
    const float* scl = (const float*)d_in[1];   // (N,3)
    const float* rot = (const float*)d_in[2];   // (N,4)
    const float* flt = (const float*)d_in[3];   // (N,3)
    const int n = in_sizes[0];                  // N

    float* cov = (float*)d_out;                 // first 9*N floats
    float* opo = cov + (size_t)9 * (size_t)n;   // then N floats

    const int blocks = (n + TILE - 1) / TILE;
    gaussian_cov_kernel<<<blocks, TILE, 0, stream>>>(opa, scl, rot, flt, cov, opo, n);
}